// SamplerBase_6322191860424
// MI455X (gfx1250) — compile-verified
//
#include <hip/hip_runtime.h>

// ---------------- problem constants ----------------
#define ROWS        256
#define COLS        128000
#define TOPK        50
#define TOPP        0.9f
#define NEGINF      (-3.402823466e+38f)
#define SEED        42ull

// ---------------- kernel geometry ------------------
#define BLOCK       256                    // 8 wave32s
#define NWAVES      (BLOCK / 32)
#define VEC         4                      // 16B per lane per tile
#define TILE_ELEMS  (BLOCK * VEC)          // 1024 floats = 4 KB tile
#define NTILES      (COLS / TILE_ELEMS)    // 125 exactly
#define HBINS       4096
#define HCOPIES     8                      // one histogram copy per wave
#define BINS_PER_T  (HBINS / BLOCK)        // 16
#define CAND_MAX    1024

static_assert(NTILES * TILE_ELEMS == COLS, "tiling must be exact");

// pointer types matching the async-load builtin: (v4i as1* src, v4i as3* dst, Ii, Ii)
typedef int v4i __attribute__((ext_vector_type(4)));
typedef __attribute__((address_space(1))) v4i* g_v4i_p;
typedef __attribute__((address_space(3))) v4i* l_v4i_p;

// order-preserving float<->uint mapping (descending float == descending key)
__device__ __forceinline__ unsigned f2o(float x) {
  unsigned u = __float_as_uint(x);
  return (u & 0x80000000u) ? ~u : (u | 0x80000000u);
}

// CDNA5 async DMA: global -> LDS, 16B per lane, tracked by ASYNCcnt
__device__ __forceinline__ void async_copy_b128(void* lds_dst, const void* gsrc) {
#if __has_builtin(__builtin_amdgcn_global_load_async_to_lds_b128)
  __builtin_amdgcn_global_load_async_to_lds_b128((g_v4i_p)gsrc, (l_v4i_p)lds_dst,
                                                 /*imm offset*/0, /*cpol*/0);
#else
  unsigned loff = (unsigned)(unsigned long long)lds_dst;  // low 32b = LDS offset
  asm volatile("global_load_async_to_lds_b128 %0, %1, off"
               :: "v"(loff), "v"(gsrc) : "memory");
#endif
}

__device__ __forceinline__ void wait_async_le1() {
#if __has_builtin(__builtin_amdgcn_s_wait_asynccnt)
  __builtin_amdgcn_s_wait_asynccnt(1);
#else
  asm volatile("s_wait_asynccnt 1" ::: "memory");
#endif
}
__device__ __forceinline__ void wait_async_le0() {
#if __has_builtin(__builtin_amdgcn_s_wait_asynccnt)
  __builtin_amdgcn_s_wait_asynccnt(0);
#else
  asm volatile("s_wait_asynccnt 0" ::: "memory");
#endif
}

// deterministic per-row uniform in [0,1) (splitmix64 hash of (seed,row))
__device__ __forceinline__ float uniform01(unsigned long long seed, unsigned row) {
  unsigned long long z = seed * 0x9E3779B97F4A7C15ull + (unsigned long long)(row + 1u);
  z = (z ^ (z >> 30)) * 0xBF58476D1CE4E5B9ull;
  z = (z ^ (z >> 27)) * 0x94D049BB133111EBull;
  z ^= (z >> 31);
  return (float)(z >> 40) * (1.0f / 16777216.0f);
}

__global__ __launch_bounds__(BLOCK)
void sampler_topk_topp_kernel(const float* __restrict__ logits, float* __restrict__ out) {
  __shared__ __align__(16) float tile[2][TILE_ELEMS];   // 8 KB double buffer
  __shared__ unsigned hist[HCOPIES][HBINS];             // 128 KB (per-wave copies)
  __shared__ unsigned chunk_sum[BLOCK];                 // 1 KB
  __shared__ float    wm[NWAVES], ws[NWAVES];           // online-softmax partials
  __shared__ float    sM, sS;
  __shared__ unsigned sthresh;
  __shared__ unsigned scand_n;
  __shared__ float    cand_v[CAND_MAX];                 // 4 KB
  __shared__ unsigned cand_i[CAND_MAX];                 // 4 KB
  __shared__ float    top_v[TOPK];
  __shared__ unsigned top_i[TOPK];
  __shared__ float    top_e[TOPK];

  const int tid  = threadIdx.x;
  const int row  = blockIdx.x;
  const int wave = tid >> 5;
  const float* rowp = logits + (size_t)row * COLS;

  for (int i = tid; i < HBINS; i += BLOCK)
#pragma unroll
    for (int c = 0; c < HCOPIES; ++c) hist[c][i] = 0u;
  if (tid == 0) scand_n = 0u;
  __syncthreads();

  // ---------- pass 1: online softmax (m,s) + key histogram, async double-buffered ----------
  float m = NEGINF, s = 0.f;
  async_copy_b128(&tile[0][tid * VEC], rowp + tid * VEC);
  for (int t = 0; t < NTILES; ++t) {
    if (t + 1 < NTILES) {
      async_copy_b128(&tile[(t + 1) & 1][tid * VEC],
                      rowp + (size_t)(t + 1) * TILE_ELEMS + tid * VEC);
      wait_async_le1();   // oldest (tile t) complete; lanes only read their own 16B
    } else {
      wait_async_le0();
    }
    const float4 v = *(const float4*)&tile[t & 1][tid * VEC];
    const float xs[4] = {v.x, v.y, v.z, v.w};
#pragma unroll
    for (int j = 0; j < 4; ++j) {
      const float x = xs[j];
      if (x > m) { s = s * __expf(m - x) + 1.f; m = x; }   // rare after warm-up
      else       { s += __expf(x - m); }
      atomicAdd(&hist[wave][f2o(x) >> 20], 1u);
    }
  }
  // wave32 combine of (m, s)
  for (int off = 16; off > 0; off >>= 1) {
    const float om = __shfl_xor(m, off);
    const float os = __shfl_xor(s, off);
    const float mm = fmaxf(m, om);
    s = s * __expf(m - mm) + os * __expf(om - mm);
    m = mm;
  }
  if ((tid & 31) == 0) { wm[wave] = m; ws[wave] = s; }
  __syncthreads();   // histogram atomics + wave partials visible

  // ---------- merge histogram copies + per-chunk sums (parallel) ----------
  {
    unsigned csum = 0;
#pragma unroll
    for (int j = 0; j < BINS_PER_T; ++j) {
      const int i = tid * BINS_PER_T + j;
      unsigned tot = 0;
#pragma unroll
      for (int c = 0; c < HCOPIES; ++c) tot += hist[c][i];
      hist[0][i] = tot;
      csum += tot;
    }
    chunk_sum[tid] = csum;
  }
  __syncthreads();

  if (tid == 0) {
    // combine per-wave (m, s)
    float M = wm[0], S = ws[0];
    for (int w = 1; w < NWAVES; ++w) {
      const float mm = fmaxf(M, wm[w]);
      S = S * __expf(M - mm) + ws[w] * __expf(wm[w] - mm);
      M = mm;
    }
    sM = M; sS = S;
    // largest bin b such that count(key >= b<<20) >= TOPK: coarse chunk scan, then bins
    unsigned acc = 0; int ch = BLOCK - 1;
    for (; ch > 0; --ch) { if (acc + chunk_sum[ch] >= TOPK) break; acc += chunk_sum[ch]; }
    int b = ch * BINS_PER_T + (BINS_PER_T - 1);
    for (; b > ch * BINS_PER_T; --b) { acc += hist[0][b]; if (acc >= TOPK) break; }
    sthresh = (unsigned)b << 20;
  }
  __syncthreads();

  const unsigned th = sthresh;

  // ---------- pass 2 (mostly L2-resident): gather candidates >= threshold ----------
  async_copy_b128(&tile[0][tid * VEC], rowp + tid * VEC);
  for (int t = 0; t < NTILES; ++t) {
    if (t + 1 < NTILES) {
      async_copy_b128(&tile[(t + 1) & 1][tid * VEC],
                      rowp + (size_t)(t + 1) * TILE_ELEMS + tid * VEC);
      wait_async_le1();
    } else {
      wait_async_le0();
    }
    const float4 v = *(const float4*)&tile[t & 1][tid * VEC];
    const float xs[4] = {v.x, v.y, v.z, v.w};
#pragma unroll
    for (int j = 0; j < 4; ++j) {
      const float x = xs[j];
      if (f2o(x) >= th) {
        unsigned pos = atomicAdd(&scand_n, 1u);
        if (pos < CAND_MAX) {
          cand_v[pos] = x;
          cand_i[pos] = (unsigned)(t * TILE_ELEMS + tid * VEC + j);
        }
      }
    }
  }
  __syncthreads();

  if (tid >= 32) return;   // wave 0 finishes the row

  // ---------- top-TOPK extraction over candidates (wave32 argmax rounds) ----------
  const int lane = tid;
  const int nc = (int)(scand_n < CAND_MAX ? scand_n : CAND_MAX);
  for (int k = 0; k < TOPK; ++k) {
    float bv = NEGINF; unsigned bi = 0xFFFFFFFFu; int bp = -1;
    for (int i = lane; i < nc; i += 32) {
      const float v = cand_v[i]; const unsigned ci = cand_i[i];
      if (v > bv || (v == bv && ci < bi)) { bv = v; bi = ci; bp = i; }
    }
    for (int off = 16; off > 0; off >>= 1) {
      const float    ov = __shfl_xor(bv, off);
      const unsigned oi = __shfl_xor(bi, off);
      const int      op = __shfl_xor(bp, off);
      if (ov > bv || (ov == bv && oi < bi)) { bv = ov; bi = oi; bp = op; }
    }
    if (lane == 0) {
      top_v[k] = bv; top_i[k] = bi;
      if (bp >= 0) cand_v[bp] = NEGINF;   // consume winner
    }
    __builtin_amdgcn_wave_barrier();      // keep DS store/loads ordered in the wave
  }

  // ---------- top-p prefix, renormalize, sample, confidence ----------
  if (tid == 0) {
    const float M = sM;
    const float S = sS;
    float cum = 0.f, Skept = 0.f;
    int nkeep = 0;
    for (int k = 0; k < TOPK; ++k) {
      const float e = __expf(top_v[k] - M);
      const float p = e / S;
      if (cum <= TOPP) { top_e[nkeep] = e; ++nkeep; Skept += e; }
      else break;       // cum is monotone: kept set is a prefix
      cum += p;
    }
    const float u = uniform01(SEED, (unsigned)row);
    const float target = u * Skept;
    float acc = 0.f; int sel = nkeep - 1;
    for (int k = 0; k < nkeep; ++k) {
      acc += top_e[k];
      if (acc >= target) { sel = k; break; }
    }
    const float conf = top_e[sel] / Skept;
    out[row]            = conf;                      // confidence
    out[ROWS + row]     = (float)top_i[sel];         // x0 (exact for idx < 2^24)
    out[2 * ROWS + row] = conf;                      // initial_confidence
  }
}

extern "C" void kernel_launch(void* const* d_in, const int* in_sizes, int n_in,
                              void* d_out, int out_size, void* d_ws, size_t ws_size,
                              hipStream_t stream) {
  (void)in_sizes; (void)n_in; (void)d_ws; (void)ws_size; (void)out_size;
  const float* logits = (const float*)d_in[0];
  float* out = (float*)d_out;
  sampler_topk_topp_kernel<<<dim3(ROWS), dim3(BLOCK), 0, stream>>>(logits, out);
}